// PreconditionedConjugateGradientModel_3083786518608
// MI455X (gfx1250) — compile-verified
//
#include <hip/hip_runtime.h>
#include <stdint.h>

static constexpr int kH = 256, kW = 256, kZ = 256;
static constexpr int kBlocks = kH * kW;            // one z-line per block
static constexpr int kN = kH * kW * kZ;
static constexpr int kIters = 4;
#define EPS 1e-6f

typedef float v2f_t __attribute__((ext_vector_type(2)));
typedef float v4f_t __attribute__((ext_vector_type(4)));
typedef float v8f_t __attribute__((ext_vector_type(8)));

// ---------------------------------------------------------------------------
// CDNA5 async global->LDS copies (gfx1250). Tracked with ASYNCcnt; LDS
// address is a VGPR byte offset (low 32 bits of the generic pointer to a
// __shared__ object). GVS mode: SADDR(base SGPR pair) + VADDR(32-bit offset).
// ---------------------------------------------------------------------------
__device__ __forceinline__ void async_copy_b128(uint32_t lds_off_bytes,
                                                uint32_t g_off_bytes,
                                                const float* base) {
#if defined(__gfx1250__)
  asm volatile("global_load_async_to_lds_b128 %0, %1, %2"
               :: "v"(lds_off_bytes), "v"(g_off_bytes), "s"(base)
               : "memory");
#else
  (void)lds_off_bytes; (void)g_off_bytes; (void)base;
#endif
}

__device__ __forceinline__ void async_wait_all() {
#if defined(__gfx1250__)
  asm volatile("s_wait_asynccnt 0" ::: "memory");
#endif
}

// ---------------------------------------------------------------------------
// Stage the 5 z-lines needed by the 7-point stencil for block `blk` (one
// (h,w) column) into tile[5][kZ] with B128 async transfers (16 B per lane):
//   line 0 = center (h,w), 1 = west (w-1), 2 = east (w+1),
//   line 3 = north (h-1),  4 = south (h+1). Out-of-domain lines are zeroed.
// Threads 0..255: thread t stages chunk (t&63) of line (t>>6); threads 0..63
// additionally stage line 4. All branches are wave-uniform.
// ---------------------------------------------------------------------------
__device__ __forceinline__ void stage_lines(const float* __restrict__ src,
                                            int blk, float* tile) {
  const int t = threadIdx.x;
  const int w = blk & (kW - 1);
  const int h = blk >> 8;
  const uint32_t ldsBase = (uint32_t)(uintptr_t)(&tile[0]);
  const uint32_t centerB = (uint32_t)blk * 1024u;      // 1 KB per z-line
  const int chunkIdx = t & 63;
  const uint32_t chunk = (uint32_t)chunkIdx * 16u;
  const int l = t >> 6;                                // line 0..3

  int32_t rel = 0;
  bool valid = true;
  if (l == 1)      { rel = -1024;   valid = (w > 0); }
  else if (l == 2) { rel = 1024;    valid = (w < kW - 1); }
  else if (l == 3) { rel = -262144; valid = (h > 0); }

  if (valid) {
    async_copy_b128(ldsBase + (uint32_t)l * 1024u + chunk,
                    (uint32_t)((int32_t)(centerB + chunk) + rel), src);
  } else {
    v4f_t zz = {};
    *(v4f_t*)&tile[l * kZ + chunkIdx * 4] = zz;
  }
  if (t < 64) {                                        // line 4 = south
    if (h < kH - 1) {
      async_copy_b128(ldsBase + 4096u + chunk, centerB + 262144u + chunk, src);
    } else {
      v4f_t zz = {};
      *(v4f_t*)&tile[4 * kZ + chunkIdx * 4] = zz;
    }
  }
  async_wait_all();
  __syncthreads();
}

// 7-point stencil from staged LDS lines; diagonal = 6 + w, Dirichlet zeros.
__device__ __forceinline__ float stencil_from_tile(const float* tile, int w) {
  const int z = threadIdx.x;
  const float c = 6.0f + (float)w;
  const float xc    = tile[z];
  const float west  = tile[1 * kZ + z];
  const float east  = tile[2 * kZ + z];
  const float north = tile[3 * kZ + z];
  const float south = tile[4 * kZ + z];
  const float bot   = (z > 0)      ? tile[z - 1] : 0.0f;
  const float top   = (z < kZ - 1) ? tile[z + 1] : 0.0f;
  return c * xc - west - east - north - south - bot - top;
}

__device__ __forceinline__ float block_sum(float v, float* red) {
  const int t = threadIdx.x;
  red[t] = v;
  __syncthreads();
  for (int s = 128; s > 0; s >>= 1) {
    if (t < s) red[t] += red[t + s];
    __syncthreads();
  }
  const float r = red[0];
  __syncthreads();
  return r;
}

// ---------------------------------------------------------------------------
// WMMA-based sum of n floats (n % 512 == 0), 256-thread single block.
// A = ones(16x4) so each V_WMMA_F32_16X16X4_F32 folds 64 new values of B
// into the f32 accumulator: D[m][n] = sum_k B[k][n] + C[m][n]. Every entry
// of the final 16x16 D is a column sum, and each column sum appears 16
// times, so total = (sum of all 256 accumulator entries) / 16 (exact).
// Valid on thread 0 only. red must hold >= 8 floats.
// ---------------------------------------------------------------------------
__device__ __forceinline__ float wmma_block_sum(const float* __restrict__ data,
                                                int n, float* red) {
  const int t = threadIdx.x;
  const int wave = t >> 5;
  const int lane = t & 31;
  v8f_t c = {};
  v2f_t a;
  a.x = 1.0f; a.y = 1.0f;
  for (int i = wave * 64; i < n; i += 512) {
    v2f_t b;
    b.x = data[i + lane];
    b.y = data[i + 32 + lane];
    c = __builtin_amdgcn_wmma_f32_16x16x4_f32(false, a, false, b,
                                              (short)0, c, false, false);
  }
  float local = c[0] + c[1] + c[2] + c[3] + c[4] + c[5] + c[6] + c[7];
  for (int off = 16; off > 0; off >>= 1) local += __shfl_down(local, off, 32);
  if (lane == 0) red[wave] = local;
  __syncthreads();
  float tot = 0.0f;
  if (t == 0) {
    for (int i = 0; i < 8; ++i) tot += red[i];
    tot *= (1.0f / 16.0f);
  }
  __syncthreads();
  return tot;
}

// ---------------------------------------------------------------------------
// r = b - A x ; p = z = M^-1 r ; partials of rho = r.z and xi = r.r
// Also copies x into the output buffer (x already staged in LDS).
// ---------------------------------------------------------------------------
__global__ void __launch_bounds__(kZ)
k_init(const float* __restrict__ x, const float* __restrict__ b,
       float* __restrict__ r, float* __restrict__ p, float* __restrict__ xout,
       float* __restrict__ partRho, float* __restrict__ partXi) {
  __shared__ alignas(16) float tile[5 * kZ];
  __shared__ float redA[kZ];
  __shared__ float redB[kZ];
  const int blk = blockIdx.x;
  const int w = blk & (kW - 1);
  const int idx = blk * kZ + threadIdx.x;
  stage_lines(x, blk, tile);
  const float st = stencil_from_tile(tile, w);
  const float rv = b[idx] - st;
  const float inv = 1.0f / (6.0f + (float)w);
  const float zv = inv * rv;
  r[idx] = rv;
  p[idx] = zv;
  xout[idx] = tile[threadIdx.x];
  const float sr = block_sum(rv * zv, redA);
  const float sx = block_sum(rv * rv, redB);
  if (threadIdx.x == 0) { partRho[blk] = sr; partXi[blk] = sx; }
}

// eta partials: p . (A p), fused — A p is never written to memory.
__global__ void __launch_bounds__(kZ)
k_eta(const float* __restrict__ p, float* __restrict__ partEta) {
  __shared__ alignas(16) float tile[5 * kZ];
  __shared__ float red[kZ];
  const int blk = blockIdx.x;
  const int w = blk & (kW - 1);
  stage_lines(p, blk, tile);
  const float st = stencil_from_tile(tile, w);
  const float pv = tile[threadIdx.x];
  const float s = block_sum(pv * st, red);
  if (threadIdx.x == 0) partEta[blk] = s;
}

// x += alpha p ; r -= alpha (A p) [recomputed from L2-resident p];
// partials of rho' = r.(M^-1 r) and xi = r.r
__global__ void __launch_bounds__(kZ)
k_update(const float* __restrict__ scal, float* __restrict__ xout,
         float* __restrict__ r, const float* __restrict__ p,
         float* __restrict__ partRho, float* __restrict__ partXi) {
  __shared__ alignas(16) float tile[5 * kZ];
  __shared__ float redA[kZ];
  __shared__ float redB[kZ];
  const int blk = blockIdx.x;
  const int w = blk & (kW - 1);
  const int idx = blk * kZ + threadIdx.x;
  const float alpha = scal[2];
  stage_lines(p, blk, tile);
  const float wv = stencil_from_tile(tile, w);
  const float pv = tile[threadIdx.x];
  xout[idx] += alpha * pv;
  const float rv = r[idx] - alpha * wv;
  r[idx] = rv;
  const float inv = 1.0f / (6.0f + (float)w);
  const float sr = block_sum(rv * rv * inv, redA);
  const float sx = block_sum(rv * rv, redB);
  if (threadIdx.x == 0) { partRho[blk] = sr; partXi[blk] = sx; }
}

// p = M^-1 r + beta p
__global__ void __launch_bounds__(kZ)
k_pupd(const float* __restrict__ scal, const float* __restrict__ r,
       float* __restrict__ p) {
  const int blk = blockIdx.x;
  const int w = blk & (kW - 1);
  const int idx = blk * kZ + threadIdx.x;
  const float beta = scal[3];
  const float inv = 1.0f / (6.0f + (float)w);
  p[idx] = inv * r[idx] + beta * p[idx];
}

// diff = x - ref : partial sum of diff^2 and partial max |diff|
__global__ void __launch_bounds__(kZ)
k_final(const float* __restrict__ xout, const float* __restrict__ ref,
        float* __restrict__ partLoss, float* __restrict__ partMax) {
  __shared__ float redA[kZ];
  __shared__ float redB[kZ];
  const int idx = blockIdx.x * kZ + threadIdx.x;
  const float d = xout[idx] - ref[idx];
  const float s = block_sum(d * d, redA);
  const int t = threadIdx.x;
  redB[t] = fabsf(d);
  __syncthreads();
  for (int sh = 128; sh > 0; sh >>= 1) {
    if (t < sh) redB[t] = fmaxf(redB[t], redB[t + sh]);
    __syncthreads();
  }
  if (t == 0) { partLoss[blockIdx.x] = s; partMax[blockIdx.x] = redB[0]; }
}

// ------------------- single-block combiners (WMMA sums) --------------------
__global__ void __launch_bounds__(256)
k_reduce_init(const float* __restrict__ pr, const float* __restrict__ px,
              float* __restrict__ scal) {
  __shared__ float red[8];
  const float sa = wmma_block_sum(pr, kBlocks, red);
  const float sb = wmma_block_sum(px, kBlocks, red);
  if (threadIdx.x == 0) { scal[0] = sa; scal[1] = sb; }   // rho, xi
}

__global__ void __launch_bounds__(256)
k_alpha(const float* __restrict__ pe, float* __restrict__ scal) {
  __shared__ float red[8];
  const float eta = wmma_block_sum(pe, kBlocks, red);
  if (threadIdx.x == 0) scal[2] = scal[0] / (eta + EPS);  // alpha
}

__global__ void __launch_bounds__(256)
k_beta(const float* __restrict__ pr, const float* __restrict__ px,
       float* __restrict__ scal) {
  __shared__ float red[8];
  const float rhoN = wmma_block_sum(pr, kBlocks, red);
  const float xi = wmma_block_sum(px, kBlocks, red);
  if (threadIdx.x == 0) {
    scal[3] = rhoN / (scal[0] + EPS);   // beta
    scal[0] = rhoN;                     // rho = rho_next
    scal[1] = xi;                       // xi
  }
}

__global__ void __launch_bounds__(256)
k_finish(const float* __restrict__ pl, const float* __restrict__ pm,
         const float* __restrict__ scal, float* __restrict__ loss,
         float* __restrict__ maxe, float* __restrict__ res) {
  __shared__ float red[8];
  __shared__ float redB[256];
  const float s = wmma_block_sum(pl, kBlocks, red);
  const int t = threadIdx.x;
  float m = 0.0f;
  for (int i = t; i < kBlocks; i += 256) m = fmaxf(m, pm[i]);
  redB[t] = m;
  __syncthreads();
  for (int sh = 128; sh > 0; sh >>= 1) {
    if (t < sh) redB[t] = fmaxf(redB[t], redB[t + sh]);
    __syncthreads();
  }
  if (t == 0) {
    *loss = s / (float)kN;
    *maxe = redB[0];
    *res = scal[1];
  }
}

// ---------------------------------------------------------------------------
extern "C" void kernel_launch(void* const* d_in, const int* in_sizes, int n_in,
                              void* d_out, int out_size, void* d_ws,
                              size_t ws_size, hipStream_t stream) {
  const float* x   = (const float*)d_in[0];
  const float* b   = (const float*)d_in[1];
  const float* ref = (const float*)d_in[2];

  float* out  = (float*)d_out;
  float* loss = out;                 // out[0]
  float* xout = out + 1;             // out[1 .. N]
  float* maxe = out + 1 + kN;        // out[N+1]
  float* res  = out + 2 + kN;        // out[N+2]

  char* ws = (char*)d_ws;
  float* r     = (float*)(ws);
  float* p     = (float*)(ws + (size_t)kN * 4);
  float* partA = (float*)(ws + (size_t)kN * 8);
  float* partB = (float*)(ws + (size_t)kN * 8 + (size_t)kBlocks * 4);
  float* scal  = (float*)(ws + (size_t)kN * 8 + (size_t)kBlocks * 8);

  dim3 grid(kBlocks), blkd(kZ);
  k_init<<<grid, blkd, 0, stream>>>(x, b, r, p, xout, partA, partB);
  k_reduce_init<<<1, 256, 0, stream>>>(partA, partB, scal);
  for (int it = 0; it < kIters; ++it) {
    k_eta<<<grid, blkd, 0, stream>>>(p, partA);
    k_alpha<<<1, 256, 0, stream>>>(partA, scal);
    k_update<<<grid, blkd, 0, stream>>>(scal, xout, r, p, partA, partB);
    k_beta<<<1, 256, 0, stream>>>(partA, partB, scal);
    k_pupd<<<grid, blkd, 0, stream>>>(scal, r, p);
  }
  k_final<<<grid, blkd, 0, stream>>>(xout, ref, partA, partB);
  k_finish<<<1, 256, 0, stream>>>(partA, partB, scal, loss, maxe, res);
}